// MultiWorldSSM_37014028157387
// MI455X (gfx1250) — compile-verified
//
#include <hip/hip_runtime.h>
#include <hip/hip_bf16.h>

// ---------------------------------------------------------------------------
// MultiWorldSSM for MI455X (gfx1250).
// - Everything except the serial h_{t-1}@Ad recurrence is dense f32 WMMA GEMM
//   (v_wmma_f32_16x16x4_f32 = exact-precision matrix pipe).
// - expm via 10-term Taylor (||0.1*A|| ~ 0.05 -> error < 1e-16).
// - Dominant cost: [4096,256]x[256,32000] vocab GEMM (67 GFLOP, compute bound;
//   524 MB output is only ~22us at 23.3 TB/s).
// - GEMM kernel: block = 8 waves; A strip (16 x K) staged once in LDS (padded,
//   conflict-free b64 reads); wave tile 16x64 (4 accumulators reuse one A
//   fragment); K-loop software-pipelined with RUNNING POINTERS (no per-step
//   64-bit multiplies); no divergent guards around WMMA (N % 64 == 0 at every
//   call site) so EXEC stays all-ones.
// ---------------------------------------------------------------------------

typedef float v2f __attribute__((ext_vector_type(2)));
typedef float v8f __attribute__((ext_vector_type(8)));

#define DM    256
#define SEQ   1024
#define BATCH 4
#define ROWS  (BATCH * SEQ)   // 4096
#define RH    64
#define NW    4

// ---------------------------------------------------------------------------
// 1) Embedding gather: xe[row, d] = emb[x[row], d]
// ---------------------------------------------------------------------------
__global__ __launch_bounds__(DM) void k_gather(const int* __restrict__ x,
                                               const float* __restrict__ emb,
                                               float* __restrict__ xe) {
    const int row = blockIdx.x;
    const int d   = threadIdx.x;
    const int tok = x[row];
    xe[(size_t)row * DM + d] = emb[(size_t)tok * DM + d];
}

// ---------------------------------------------------------------------------
// 2) Router: mean-pool -> relu(.@rw1+rb1) -> .@rw2+rb2 -> softmax.
//    Writes wT[i*4 + b] (world-major) so GEMM epilogue can index by batch.
// ---------------------------------------------------------------------------
__global__ __launch_bounds__(DM) void k_router(const float* __restrict__ xe,
                                               const float* __restrict__ rw1,
                                               const float* __restrict__ rb1,
                                               const float* __restrict__ rw2,
                                               const float* __restrict__ rb2,
                                               float* __restrict__ wT) {
    __shared__ float m[DM];
    __shared__ float r[RH];
    __shared__ float lg[NW];
    const int b = blockIdx.x;
    const int t = threadIdx.x;

    float s = 0.f;
    const float* base = xe + (size_t)b * SEQ * DM + t;
    for (int i = 0; i < SEQ; ++i) s += base[(size_t)i * DM];
    m[t] = s * (1.0f / SEQ);
    __syncthreads();

    if (t < RH) {
        float h = rb1[t];
        for (int d = 0; d < DM; ++d) h += m[d] * rw1[d * RH + t];
        r[t] = h > 0.f ? h : 0.f;
    }
    __syncthreads();

    if (t < NW) {
        float h = rb2[t];
        for (int j = 0; j < RH; ++j) h += r[j] * rw2[j * NW + t];
        lg[t] = h;
    }
    __syncthreads();

    if (t == 0) {
        float mx = lg[0];
        for (int i = 1; i < NW; ++i) mx = lg[i] > mx ? lg[i] : mx;
        float e[NW], sum = 0.f;
        for (int i = 0; i < NW; ++i) { e[i] = __expf(lg[i] - mx); sum += e[i]; }
        const float inv = 1.0f / sum;
        for (int i = 0; i < NW; ++i) wT[i * BATCH + b] = e[i] * inv;
    }
}

// ---------------------------------------------------------------------------
// 3) expm via Taylor:  M = 0.1*A; P = I; T = I;  then T += (P @= M/k), k=1..10
// ---------------------------------------------------------------------------
__global__ __launch_bounds__(256) void k_expm_init(const float* __restrict__ A,
                                                   float* __restrict__ M,
                                                   float* __restrict__ P,
                                                   float* __restrict__ T, int ds) {
    const int idx = blockIdx.x * 256 + threadIdx.x;
    if (idx >= ds * ds) return;
    M[idx] = A[idx] * 0.1f;
    const float id = ((idx / ds) == (idx % ds)) ? 1.0f : 0.0f;
    P[idx] = id;
    T[idx] = id;
}

__global__ __launch_bounds__(256) void k_mm_taylor(const float* __restrict__ P,
                                                   const float* __restrict__ M,
                                                   float* __restrict__ Pn,
                                                   float* __restrict__ T,
                                                   int ds, float invk) {
    const int col = blockIdx.x * 16 + threadIdx.x;
    const int row = blockIdx.y * 16 + threadIdx.y;
    float s = 0.f;
    const float* pr = P + (size_t)row * ds;
    const float* mc = M + col;
    for (int k = 0; k < ds; ++k) { s += pr[k] * (*mc); mc += ds; }
    s *= invk;
    const size_t idx = (size_t)row * ds + col;
    Pn[idx] = s;
    T[idx] += s;
}

// ---------------------------------------------------------------------------
// 4) WMMA f32 GEMM:  out[M,N] = rowScale(row) * (A[M,K] @ B[K,N]) (+ out)(+ bias)
//    Requirements: M%16==0, N%64==0, K%4==0 (all call sites satisfy this).
// ---------------------------------------------------------------------------
__global__ __launch_bounds__(256)
void k_wmma_gemm_f32(const float* __restrict__ A, const float* __restrict__ B,
                     float* __restrict__ out, int Ndim, int Kdim,
                     const float* __restrict__ rowScale,  // [4] by batch or null
                     const float* __restrict__ bias,      // [N] or null
                     int accum) {
    extern __shared__ float shA[];            // 16 x (Kdim + 4), padded
    const int ldA = Kdim + 4;
    const int m0  = blockIdx.y << 4;

    // --- Stage the contiguous 16xK A strip into LDS (coalesced b128 loads) ---
    {
        const float4* Ag = reinterpret_cast<const float4*>(A + (size_t)m0 * Kdim);
        const int nf4 = (16 * Kdim) >> 2;
        for (int i = threadIdx.x; i < nf4; i += 256) {
            const int e   = i << 2;
            const int row = e / Kdim;
            const int col = e - row * Kdim;
            *reinterpret_cast<float4*>(&shA[row * ldA + col]) = Ag[i];
        }
    }
    __syncthreads();

    const int lane  = threadIdx.x & 31;
    const int wave  = threadIdx.x >> 5;
    const int nBase = (blockIdx.x << 9) + (wave << 6);
    if (nBase >= Ndim) return;                // wave-uniform; no barriers follow

    const int lmod    = lane & 15;
    const int kHalf   = (lane >> 4) << 1;     // 0 or 2
    const int colBase = nBase + lmod;

    // Running pointers: one LDS pointer (+4 floats/step) and one global
    // pointer (+4*Ndim floats/step). No per-iteration 64-bit multiplies.
    const float*  aPtr  = shA + lmod * ldA + kHalf;
    const float*  Bp    = B + (size_t)kHalf * Ndim + colBase;
    const size_t  bRow  = (size_t)Ndim;       // one K-row, in floats
    const size_t  bStep = (size_t)4 * Ndim;   // advance per k-group

    v8f acc0 = {}, acc1 = {}, acc2 = {}, acc3 = {};

    // --- Software-pipelined K loop: prefetch k+4 while issuing 4 WMMAs ------
    v2f a, b0, b1, b2, b3;
    {
        a.x = aPtr[0]; a.y = aPtr[1];
        const float* Bq = Bp + bRow;
        b0.x = Bp[0];  b0.y = Bq[0];
        b1.x = Bp[16]; b1.y = Bq[16];
        b2.x = Bp[32]; b2.y = Bq[32];
        b3.x = Bp[48]; b3.y = Bq[48];
    }
    for (int k = 0; k + 4 < Kdim; k += 4) {
        aPtr += 4;
        Bp   += bStep;
        v2f an, bn0, bn1, bn2, bn3;
        an.x = aPtr[0]; an.y = aPtr[1];
        const float* Bq = Bp + bRow;
        bn0.x = Bp[0];  bn0.y = Bq[0];
        bn1.x = Bp[16]; bn1.y = Bq[16];
        bn2.x = Bp[32]; bn2.y = Bq[32];
        bn3.x = Bp[48]; bn3.y = Bq[48];

        acc0 = __builtin_amdgcn_wmma_f32_16x16x4_f32(false, a, false, b0, (short)0, acc0, false, false);
        acc1 = __builtin_amdgcn_wmma_f32_16x16x4_f32(false, a, false, b1, (short)0, acc1, false, false);
        acc2 = __builtin_amdgcn_wmma_f32_16x16x4_f32(false, a, false, b2, (short)0, acc2, false, false);
        acc3 = __builtin_amdgcn_wmma_f32_16x16x4_f32(false, a, false, b3, (short)0, acc3, false, false);

        a = an; b0 = bn0; b1 = bn1; b2 = bn2; b3 = bn3;
    }
    acc0 = __builtin_amdgcn_wmma_f32_16x16x4_f32(false, a, false, b0, (short)0, acc0, false, false);
    acc1 = __builtin_amdgcn_wmma_f32_16x16x4_f32(false, a, false, b1, (short)0, acc1, false, false);
    acc2 = __builtin_amdgcn_wmma_f32_16x16x4_f32(false, a, false, b2, (short)0, acc2, false, false);
    acc3 = __builtin_amdgcn_wmma_f32_16x16x4_f32(false, a, false, b3, (short)0, acc3, false, false);

    // --- Epilogue: C/D layout row = m0 + r + 8*(lane>=16), col = +t*16 ------
    const int rowBase = m0 + ((lane >> 4) << 3);
#pragma unroll
    for (int t = 0; t < 4; ++t) {
        const v8f acc = (t == 0) ? acc0 : (t == 1) ? acc1 : (t == 2) ? acc2 : acc3;
        const int col = colBase + t * 16;
        const float bv = bias ? bias[col] : 0.0f;
        float* op = out + (size_t)rowBase * Ndim + col;
#pragma unroll
        for (int r = 0; r < 8; ++r) {
            const int row = rowBase + r;
            float val = acc[r];
            if (rowScale) val *= rowScale[row >> 10];   // batch = row / SEQ
            if (accum) val += *op;
            *op = val + bv;
            op += Ndim;
        }
    }
}

// ---------------------------------------------------------------------------
// 5) Serial recurrence (in place on U): h_t = h_{t-1} @ Ad + u_t.
//    One block per batch, ds threads; h lives in LDS, Ad streams from L2.
// ---------------------------------------------------------------------------
__global__ void k_scan(const float* __restrict__ Ad, float* __restrict__ UH, int ds) {
    extern __shared__ float sh[];   // ds floats: h_{t-1}
    const int b = blockIdx.x;
    const int j = threadIdx.x;
    float* base = UH + (size_t)b * SEQ * ds;

    sh[j] = 0.0f;
    __syncthreads();

    for (int t = 0; t < SEQ; ++t) {
        float s = base[(size_t)t * ds + j];
        const float* adj = Ad + j;
        for (int k = 0; k + 4 <= ds; k += 4) {
            s += sh[k + 0] * adj[0];
            s += sh[k + 1] * adj[(size_t)ds];
            s += sh[k + 2] * adj[(size_t)2 * ds];
            s += sh[k + 3] * adj[(size_t)3 * ds];
            adj += (size_t)4 * ds;
        }
        __syncthreads();            // all reads of h_{t-1} done
        sh[j] = s;
        base[(size_t)t * ds + j] = s;
        __syncthreads();            // h_t visible before next step
    }
}

// ---------------------------------------------------------------------------
// 6) mix[row,d] = xe[row,d] * sum_i wT[i,b] * D_i[d]   (the skip/D term)
// ---------------------------------------------------------------------------
__global__ __launch_bounds__(DM) void k_mix_init(const float* __restrict__ xe,
                                                 const float* __restrict__ D0,
                                                 const float* __restrict__ D1,
                                                 const float* __restrict__ D2,
                                                 const float* __restrict__ D3,
                                                 const float* __restrict__ wT,
                                                 float* __restrict__ mix) {
    const int row = blockIdx.x;
    const int d   = threadIdx.x;
    const int b   = row >> 10;
    const float c = wT[0 * BATCH + b] * D0[d] + wT[1 * BATCH + b] * D1[d] +
                    wT[2 * BATCH + b] * D2[d] + wT[3 * BATCH + b] * D3[d];
    const size_t idx = (size_t)row * DM + d;
    mix[idx] = c * xe[idx];
}

// ---------------------------------------------------------------------------
// Launch
// ---------------------------------------------------------------------------
extern "C" void kernel_launch(void* const* d_in, const int* in_sizes, int n_in,
                              void* d_out, int out_size, void* d_ws, size_t ws_size,
                              hipStream_t stream) {
    (void)in_sizes; (void)n_in; (void)out_size; (void)ws_size;

    const int DS[NW] = {64, 128, 256, 512};

    const int*   x   = (const int*)d_in[0];
    const float* emb = (const float*)d_in[1];
    const float* Aw[NW], *Bw[NW], *Cw[NW], *Dw[NW];
    for (int i = 0; i < NW; ++i) {
        Aw[i] = (const float*)d_in[2 + 4 * i];
        Bw[i] = (const float*)d_in[3 + 4 * i];
        Cw[i] = (const float*)d_in[4 + 4 * i];
        Dw[i] = (const float*)d_in[5 + 4 * i];
    }
    const float* rw1 = (const float*)d_in[18];
    const float* rb1 = (const float*)d_in[19];
    const float* rw2 = (const float*)d_in[20];
    const float* rb2 = (const float*)d_in[21];
    const float* ow  = (const float*)d_in[22];
    const float* ob  = (const float*)d_in[23];
    float* out = (float*)d_out;

    // Workspace carve-up (floats). Total ~7.5M floats (~30 MB).
    float* wsf = (float*)d_ws;
    size_t off = 0;
    float* xe  = wsf + off; off += (size_t)ROWS * DM;
    float* mix = wsf + off; off += (size_t)ROWS * DM;
    float* wT  = wsf + off; off += 64;
    float *Mm[NW], *P0[NW], *P1[NW], *Tm[NW], *U[NW];
    for (int i = 0; i < NW; ++i) {
        const size_t n2 = (size_t)DS[i] * DS[i];
        Mm[i] = wsf + off; off += n2;
        P0[i] = wsf + off; off += n2;
        P1[i] = wsf + off; off += n2;
        Tm[i] = wsf + off; off += n2;
    }
    for (int i = 0; i < NW; ++i) { U[i] = wsf + off; off += (size_t)ROWS * DS[i]; }

    // 1) gather + 2) router
    k_gather<<<ROWS, DM, 0, stream>>>(x, emb, xe);
    k_router<<<BATCH, DM, 0, stream>>>(xe, rw1, rb1, rw2, rb2, wT);

    // 3) per-world: expm(A*0.1) via 10-term Taylor; 4) U = xe @ B; 5) scan
    for (int i = 0; i < NW; ++i) {
        const int ds = DS[i];
        const int n2 = ds * ds;
        k_expm_init<<<(n2 + 255) / 256, 256, 0, stream>>>(Aw[i], Mm[i], P0[i], Tm[i], ds);
        float* pc = P0[i];
        float* pn = P1[i];
        for (int k = 1; k <= 10; ++k) {
            dim3 g(ds / 16, ds / 16), blk(16, 16);
            k_mm_taylor<<<g, blk, 0, stream>>>(pc, Mm[i], pn, Tm[i], ds, 1.0f / (float)k);
            float* tmp = pc; pc = pn; pn = tmp;
        }
        // Ad now in Tm[i].
        dim3 gU((ds + 511) / 512, ROWS / 16);
        k_wmma_gemm_f32<<<gU, 256, 16 * (DM + 4) * sizeof(float), stream>>>(
            xe, Bw[i], U[i], ds, DM, nullptr, nullptr, 0);
        k_scan<<<BATCH, ds, ds * sizeof(float), stream>>>(Tm[i], U[i], ds);
    }

    // 6) mix = D-term; 7) mix += w_i * (H_i @ C_i)
    k_mix_init<<<ROWS, DM, 0, stream>>>(xe, Dw[0], Dw[1], Dw[2], Dw[3], wT, mix);
    for (int i = 0; i < NW; ++i) {
        dim3 gY((DM + 511) / 512, ROWS / 16);
        k_wmma_gemm_f32<<<gY, 256, 16 * (DS[i] + 4) * sizeof(float), stream>>>(
            U[i], Cw[i], mix, DM, DS[i], wT + BATCH * i, nullptr, 1);
    }

    // 8) logits = mix @ ow + ob   (the 67-GFLOP dominant GEMM)
    {
        const int N = 32000;
        dim3 gO((N + 511) / 512, ROWS / 16);
        k_wmma_gemm_f32<<<gO, 256, 16 * (DM + 4) * sizeof(float), stream>>>(
            mix, ow, out, N, DM, nullptr, ob, 0);
    }
}